// WarpCTCLoss_84610855731591
// MI455X (gfx1250) — compile-verified
//
#include <hip/hip_runtime.h>
#include <math.h>

// ---------------- problem constants (from reference setup_inputs) -----------
#define CB      32          // batch
#define CT      1024        // time steps
#define CV      1024        // vocab
#define CL      128         // max target length
#define CS      257         // 2*CL + 1 extended states
#define NLAB    129         // blank + CL target labels per batch row
#define PAD     132         // padded gather stride (multiple of 4 -> 16B aligned)
#define NEGF    (-1e30f)

// ---------------- CDNA5 feature gates (device pass only) --------------------
#if defined(__HIP_DEVICE_COMPILE__) && __has_builtin(__builtin_amdgcn_global_load_async_to_lds_b128)
#define HAS_ASYNC 1
#warning "CDNA5 gate: using __builtin_amdgcn_global_load_async_to_lds_b128"
#else
#define HAS_ASYNC 0
#if defined(__HIP_DEVICE_COMPILE__)
#warning "CDNA5 gate: async-to-LDS builtin NOT available, using direct-copy fallback"
#endif
#endif

#if defined(__HIP_DEVICE_COMPILE__) && __has_builtin(__builtin_amdgcn_wmma_f32_16x16x4_f32)
#define HAS_WMMA4 1
#warning "CDNA5 gate: using __builtin_amdgcn_wmma_f32_16x16x4_f32"
#else
#define HAS_WMMA4 0
#if defined(__HIP_DEVICE_COMPILE__)
#warning "CDNA5 gate: wmma f32 16x16x4 builtin NOT available, using shfl fallback"
#endif
#endif

#if HAS_ASYNC
#define GAS __attribute__((address_space(1)))
#define LAS __attribute__((address_space(3)))
typedef int v4i_vs __attribute__((vector_size(16)));   // 16-byte payload type
#endif

typedef __attribute__((ext_vector_type(2))) float v2f;
typedef __attribute__((ext_vector_type(8))) float v8f;

__device__ inline void wait_async_copies() {
#if HAS_ASYNC
#if __has_builtin(__builtin_amdgcn_s_wait_asynccnt)
  __builtin_amdgcn_s_wait_asynccnt(0);
#else
  asm volatile("s_wait_asynccnt 0" ::: "memory");
#endif
#endif
}

// 16B async (or direct) copy, per lane
__device__ inline void copy16(float* dst_lds, const float* src_g) {
#if HAS_ASYNC
  __builtin_amdgcn_global_load_async_to_lds_b128(
      (GAS v4i_vs*)src_g, (LAS v4i_vs*)dst_lds, 0, 0);
#else
  *(float4*)dst_lds = *(const float4*)src_g;
#endif
}

// -------- wave32 reductions -------------------------------------------------
__device__ inline float wave_max32(float x) {
  for (int off = 16; off > 0; off >>= 1)
    x = fmaxf(x, __shfl_xor(x, off, 32));
  return x;
}

__device__ inline float wave_sum32(float x) {
#if HAS_WMMA4
  // A (16x4 f32): VGPR0 = x  -> A[m,0]=x[lane m], A[m,2]=x[lane m+16]
  //               VGPR1 = 0  -> A[m,1]=A[m,3]=0
  // B (4x16 f32): all ones.  D[m,n] = x[m] + x[m+16]  (exact fma adds)
  v2f a; a[0] = x;    a[1] = 0.0f;
  v2f b; b[0] = 1.0f; b[1] = 1.0f;
  v8f c = {};
  c = __builtin_amdgcn_wmma_f32_16x16x4_f32(false, a, false, b,
                                            (short)0, c, false, false);
  // VGPR j holds M=j (lanes 0-15) and M=j+8 (lanes 16-31); columns identical.
  float e = ((c[0] + c[1]) + (c[2] + c[3])) + ((c[4] + c[5]) + (c[6] + c[7]));
  return e + __shfl_xor(e, 16, 32);   // halves: sum(M 0..7) + sum(M 8..15)
#else
  for (int off = 16; off > 0; off >>= 1)
    x += __shfl_xor(x, off, 32);
  return x;
#endif
}

// ============================================================================
// Phase 1: per row (b,t): lse = logsumexp over V; emit gathered
//          lp[b,t,j] = logits[b,t,lab_j] - lse for j in [0,129)
// grid = B*T/8 blocks, 256 threads (8 waves, one wave per 4KB row)
// ============================================================================
__global__ void __launch_bounds__(256)
ctc_row_lse_gather(const float* __restrict__ logits,
                   const int*   __restrict__ targets,
                   float*       __restrict__ gath) {
  __shared__ __align__(16) float stage[8][CV];   // 32 KB
  __shared__ int lab[NLAB];

  const int tid  = threadIdx.x;
  const int w    = tid >> 5;
  const int lane = tid & 31;
  const int row0 = blockIdx.x * 8;
  const int b    = row0 >> 10;                   // CT == 1024
  const int row  = row0 + w;

  if (tid < CL) lab[tid + 1] = targets[b * CL + tid];
  if (tid == 0) lab[0] = 0;                      // blank

  const float* src = logits + (size_t)row * CV;

  // ---- stage the 4KB row into LDS (async on CDNA5) ----
  for (int i = 0; i < 8; ++i) {
    const int off = i * 128 + lane * 4;          // floats
    copy16(&stage[w][off], src + off);
  }
  wait_async_copies();

  // ---- lane-local max over 32 values, then wave max ----
  float vals[32];
  float m = -INFINITY;
  for (int i = 0; i < 8; ++i) {
    float4 v = *(float4*)&stage[w][i * 128 + lane * 4];
    vals[4 * i + 0] = v.x; vals[4 * i + 1] = v.y;
    vals[4 * i + 2] = v.z; vals[4 * i + 3] = v.w;
    m = fmaxf(m, fmaxf(fmaxf(v.x, v.y), fmaxf(v.z, v.w)));
  }
  m = wave_max32(m);

  float ssum = 0.0f;
  for (int k = 0; k < 32; ++k) ssum += __expf(vals[k] - m);
  ssum = wave_sum32(ssum);                        // WMMA-based wave reduction
  const float lse = m + __logf(ssum);

  __syncthreads();                                // lab[] visible

  // ---- gather label logits from the LDS-resident row ----
  float* grow = gath + (size_t)row * PAD;
  for (int j = lane; j < NLAB; j += 32)
    grow[j] = stage[w][lab[j]] - lse;
}

// ============================================================================
// Phase 2: CTC forward recursion. One block per batch element; thread s owns
// extended state s. lp tiles (8 timesteps) are double-buffered in LDS via the
// async-to-LDS path so the next tile streams while the current one computes.
// ============================================================================
__global__ void __launch_bounds__(288)
ctc_alpha(const float* __restrict__ gath,
          const int*   __restrict__ targets,
          const int*   __restrict__ loglen,
          const int*   __restrict__ tgtlen,
          float*       __restrict__ out) {
  constexpr int TT = 8;                           // timesteps per tile
  __shared__ __align__(16) float ltile[2][TT * PAD];  // 2 x 4224 B
  __shared__ float alpha[2][CS];
  __shared__ int lab[NLAB];

  const int b   = blockIdx.x;
  const int tid = threadIdx.x;
  const float* g = gath + (size_t)b * CT * PAD;

  if (tid < CL) lab[tid + 1] = targets[b * CL + tid];
  if (tid == 0) lab[0] = 0;
  __syncthreads();

  const int tl = tgtlen[b];
  const int Tb = loglen[b];
  const int Sv = 2 * tl + 1;
  const int s  = tid;
  const bool active = (s < CS);

  int  idx = 0;
  bool canskip = false;
  if (active && (s & 1)) {
    idx = (s >> 1) + 1;                           // gather column for label s
    canskip = (s >= 3) && (lab[idx] != lab[idx - 1]);
  }

  const int ntile = (Tb + TT - 1) / TT;

  // tile loader: 1056 floats = 264 x 16B, one 16B per thread
  #define LOAD_TILE(bufp, t0)                                        \
    do {                                                             \
      if (tid < (TT * PAD) / 4)                                      \
        copy16(&ltile[(bufp)][tid * 4], g + (size_t)(t0) * PAD + tid * 4); \
    } while (0)

  LOAD_TILE(0, 0);
  wait_async_copies();
  __syncthreads();

  // ---- t = 0 init ----
  if (active) {
    float a = NEGF;
    if (s == 0)      a = ltile[0][0];             // blank
    else if (s == 1) a = ltile[0][1];             // first label
    if (s >= Sv) a = NEGF;
    alpha[0][s] = a;
  }
  if (ntile > 1) LOAD_TILE(1, TT);                // prefetch tile 1
  __syncthreads();

  int cur = 0;
  for (int tile = 0; tile < ntile; ++tile) {
    const int buf = tile & 1;
    const float* lt = &ltile[buf][0];
    const int tlo = (tile == 0) ? 1 : tile * TT;
    const int tcap = (tile + 1) * TT;
    const int thi = (Tb < tcap) ? Tb : tcap;

    for (int t = tlo; t < thi; ++t) {
      if (active) {
        const float a1 = alpha[cur][s];
        const float a2 = (s > 0) ? alpha[cur][s - 1] : NEGF;
        const float a3 = canskip ? alpha[cur][s - 2] : NEGF;
        const float m  = fmaxf(a1, fmaxf(a2, a3));
        float nv = m + __logf(__expf(a1 - m) + __expf(a2 - m) + __expf(a3 - m))
                     + lt[(t - tile * TT) * PAD + idx];
        if (s >= Sv) nv = NEGF;
        alpha[cur ^ 1][s] = nv;
      }
      __syncthreads();
      cur ^= 1;
    }

    if (tile + 1 < ntile) {
      wait_async_copies();                        // tile+1 resident
      if (tile + 2 < ntile) LOAD_TILE(buf, (tile + 2) * TT);  // refill drained buf
      __syncthreads();
    }
  }
  #undef LOAD_TILE

  if (tid == 0) {
    const float eb = alpha[cur][2 * tl];
    const float el = alpha[cur][2 * tl - 1];
    const float m  = fmaxf(eb, el);
    out[b] = -(m + __logf(__expf(eb - m) + __expf(el - m)));
  }
}

// ============================================================================
extern "C" void kernel_launch(void* const* d_in, const int* in_sizes, int n_in,
                              void* d_out, int out_size, void* d_ws, size_t ws_size,
                              hipStream_t stream) {
  (void)in_sizes; (void)n_in; (void)out_size; (void)ws_size;
  const float* logits  = (const float*)d_in[0];   // [B,T,V] f32
  const int*   targets = (const int*)d_in[1];     // [B,L]   i32
  const int*   loglen  = (const int*)d_in[2];     // [B]     i32
  const int*   tgtlen  = (const int*)d_in[3];     // [B]     i32
  float*       out     = (float*)d_out;           // [B]     f32
  float*       gath    = (float*)d_ws;            // B*T*PAD f32 = 17.3 MB

  ctc_row_lse_gather<<<(CB * CT) / 8, 256, 0, stream>>>(logits, targets, gath);
  ctc_alpha<<<CB, 288, 0, stream>>>(gath, targets, loglen, tgtlen, out);
}